// NCC_vfa_fast_81647328297397
// MI455X (gfx1250) — compile-verified
//
#include <hip/hip_runtime.h>

// NCC loss (9^3 box-filter local normalized cross-correlation) for 192^3 f32.
// Fused single-pass tile kernel: each block produces a 16^3 output tile from a
// 24^3 zero-padded halo tile kept in LDS, doing the three separable box-filter
// passes as banded matmuls with V_WMMA_F32_16X16X4_F32 (full f32 precision).

typedef __attribute__((ext_vector_type(2))) float v2f;
typedef __attribute__((ext_vector_type(8))) float v8f;

#define NV 192
#define TI 16
#define HALO 4
#define SRC 24                    // TI + 2*HALO
#define NB 12                     // NV / TI
#define NBLK (NB * NB * NB)       // 1728
#define SRC3 (SRC * SRC * SRC)    // 13824
#define SLAB_A 30720              // floats: max(sI+sJ = 27648, buf2 = 30720)
#define SLAB_B 46080              // floats: max(buf1 = 46080, out5 = 20480)
#define LDS_FLOATS (SLAB_A + SLAB_B + 8)   // 76808 floats = 307232 B < 320 KB
#define KSTEPS 6                  // K = 24 in steps of 4
#define EPS 1.1920929e-7f         // float32 machine epsilon (jnp.finfo eps)

// 9-tap zero-padded box weight: out[n] = (1/9) * sum_{k=n}^{n+8} in[k]
__device__ __forceinline__ float band_w(int k, int n) {
    return (k >= n && k <= n + 8) ? (1.0f / 9.0f) : 0.0f;
}

__global__ __launch_bounds__(256) void ncc_main(const float* __restrict__ pred,
                                                const float* __restrict__ target,
                                                float* __restrict__ bsum)
{
    extern __shared__ float smem[];
    float* slabA = smem;                     // raw tiles, later buf2
    float* slabB = smem + SLAB_A;            // buf1, later out5
    float* red   = smem + SLAB_A + SLAB_B;   // 8-float wave-partials

    float* sI   = slabA;                     // [24][24][24]  (target)
    float* sJ   = slabA + SRC3;              // [24][24][24]  (pred)
    float* buf1 = slabB;                     // [5][24][24][16]  after W pass
    float* buf2 = slabA;                     // [5][24][16][16]  after H pass
    float* out5 = slabB;                     // [5][16][16][16]  after D pass

    const int tid  = threadIdx.x;
    const int lane = tid & 31;
    const int wave = tid >> 5;
    const int lo   = lane & 15;
    const bool hi  = lane >= 16;

    const int w0 = blockIdx.x * TI - HALO;
    const int h0 = blockIdx.y * TI - HALO;
    const int d0 = blockIdx.z * TI - HALO;

    // ---- load 24^3 halo tiles, zero-padded at the volume border ----------
    for (int i = tid; i < SRC3; i += 256) {
        int dz = i / (SRC * SRC);
        int r  = i - dz * (SRC * SRC);
        int dy = r / SRC;
        int dx = r - dy * SRC;
        int gd = d0 + dz, gh = h0 + dy, gw = w0 + dx;
        bool ok = ((unsigned)gd < NV) && ((unsigned)gh < NV) && ((unsigned)gw < NV);
        int gidx = (gd * NV + gh) * NV + gw;
        sI[i] = ok ? target[gidx] : 0.0f;
        sJ[i] = ok ? pred[gidx]   : 0.0f;
    }
    __syncthreads();

    // ---- pass 1: filter along W. lines m = dz*24+dy (576), K=24 -> N=16 --
    // Channel products (I, J, I*I, J*J, I*J) are formed on the fly.
    for (int job = wave; job < 180; job += 8) {     // 36 m-tiles x 5 channels
        int c = job % 5;
        int t = job / 5;
        v8f acc = {};
        int base = (t * 16 + lo) * SRC;             // A-row base in sI/sJ
        #pragma unroll
        for (int s = 0; s < KSTEPS; ++s) {
            int k0 = 4 * s + (hi ? 2 : 0);
            float i0 = sI[base + k0],     j0 = sJ[base + k0];
            float i1 = sI[base + k0 + 1], j1 = sJ[base + k0 + 1];
            v2f a, b;
            a.x = (c == 0) ? i0 : (c == 1) ? j0 : (c == 2) ? i0 * i0
                 : (c == 3) ? j0 * j0 : i0 * j0;
            a.y = (c == 0) ? i1 : (c == 1) ? j1 : (c == 2) ? i1 * i1
                 : (c == 3) ? j1 * j1 : i1 * j1;
            b.x = band_w(k0, lo);
            b.y = band_w(k0 + 1, lo);
            acc = __builtin_amdgcn_wmma_f32_16x16x4_f32(
                      false, a, false, b, (short)0, acc, false, false);
        }
        #pragma unroll
        for (int r = 0; r < 8; ++r) {
            int m  = t * 16 + (hi ? r + 8 : r);
            int dz = m / SRC;
            int dy = m - dz * SRC;
            buf1[((c * SRC + dz) * SRC + dy) * TI + lo] = acc[r];
        }
    }
    __syncthreads();

    // ---- pass 2: filter along H. per-channel lines m = dz*16+wx (384) ----
    for (int job = wave; job < 120; job += 8) {     // 24 m-tiles x 5 channels
        int c = job % 5;
        int t = job / 5;
        v8f acc = {};
        int mA = t * 16 + lo;
        int dz = mA >> 4;
        int wx = mA & 15;
        #pragma unroll
        for (int s = 0; s < KSTEPS; ++s) {
            int k0 = 4 * s + (hi ? 2 : 0);
            v2f a, b;
            a.x = buf1[((c * SRC + dz) * SRC + k0) * TI + wx];
            a.y = buf1[((c * SRC + dz) * SRC + k0 + 1) * TI + wx];
            b.x = band_w(k0, lo);
            b.y = band_w(k0 + 1, lo);
            acc = __builtin_amdgcn_wmma_f32_16x16x4_f32(
                      false, a, false, b, (short)0, acc, false, false);
        }
        #pragma unroll
        for (int r = 0; r < 8; ++r) {
            int m   = t * 16 + (hi ? r + 8 : r);
            int mdz = m >> 4;
            int mwx = m & 15;
            buf2[((c * SRC + mdz) * TI + lo) * TI + mwx] = acc[r];
        }
    }
    __syncthreads();

    // ---- pass 3: filter along D. per-channel lines m = hy*16+wx (256) ----
    for (int job = wave; job < 80; job += 8) {      // 16 m-tiles x 5 channels
        int c = job % 5;
        int t = job / 5;
        v8f acc = {};
        int mA = t * 16 + lo;
        int hy = mA >> 4;
        int wx = mA & 15;
        #pragma unroll
        for (int s = 0; s < KSTEPS; ++s) {
            int k0 = 4 * s + (hi ? 2 : 0);
            v2f a, b;
            a.x = buf2[((c * SRC + k0) * TI + hy) * TI + wx];
            a.y = buf2[((c * SRC + k0 + 1) * TI + hy) * TI + wx];
            b.x = band_w(k0, lo);
            b.y = band_w(k0 + 1, lo);
            acc = __builtin_amdgcn_wmma_f32_16x16x4_f32(
                      false, a, false, b, (short)0, acc, false, false);
        }
        #pragma unroll
        for (int r = 0; r < 8; ++r) {
            int m   = t * 16 + (hi ? r + 8 : r);
            int hy2 = m >> 4;
            int wx2 = m & 15;
            out5[((c * TI + lo) * TI + hy2) * TI + wx2] = acc[r];
        }
    }
    __syncthreads();

    // ---- per-voxel NCC ratio + deterministic block reduction -------------
    float lsum = 0.0f;
    for (int i = tid; i < TI * TI * TI; i += 256) {
        float mu1 = out5[i];               // filtered I  (target mean)
        float mu2 = out5[4096 + i];        // filtered J  (pred mean)
        float f2  = out5[2 * 4096 + i];    // filtered I*I
        float f3  = out5[3 * 4096 + i];    // filtered J*J
        float f4  = out5[4 * 4096 + i];    // filtered I*J
        float s1  = f2 - mu1 * mu1;
        float s2  = f3 - mu2 * mu2;
        float s12 = f4 - mu1 * mu2;
        lsum += s12 * s12 / (s1 * s2 + EPS);
    }
    for (int off = 16; off > 0; off >>= 1)
        lsum += __shfl_down(lsum, off);
    if (lane == 0) red[wave] = lsum;
    __syncthreads();
    if (tid == 0) {
        float t = 0.0f;
        for (int w = 0; w < 8; ++w) t += red[w];
        bsum[(blockIdx.z * NB + blockIdx.y) * NB + blockIdx.x] = t;
    }
}

__global__ __launch_bounds__(256) void ncc_finalize(const float* __restrict__ bsum,
                                                    float* __restrict__ out)
{
    __shared__ float red[8];
    int tid = threadIdx.x;
    float s = 0.0f;
    for (int i = tid; i < NBLK; i += 256) s += bsum[i];
    for (int off = 16; off > 0; off >>= 1) s += __shfl_down(s, off);
    if ((tid & 31) == 0) red[tid >> 5] = s;
    __syncthreads();
    if (tid == 0) {
        float t = 0.0f;
        for (int w = 0; w < 8; ++w) t += red[w];
        out[0] = 1.0f - t * (1.0f / 7077888.0f);   // mean over 192^3 voxels
    }
}

extern "C" void kernel_launch(void* const* d_in, const int* in_sizes, int n_in,
                              void* d_out, int out_size, void* d_ws, size_t ws_size,
                              hipStream_t stream)
{
    const float* pred   = (const float*)d_in[0];   // J
    const float* target = (const float*)d_in[1];   // I
    float* out  = (float*)d_out;
    float* bsum = (float*)d_ws;                    // 1728 floats of scratch

    dim3 grid(NB, NB, NB);
    size_t shmem = (size_t)LDS_FLOATS * sizeof(float);
    hipLaunchKernelGGL(ncc_main, grid, dim3(256), shmem, stream, pred, target, bsum);
    hipLaunchKernelGGL(ncc_finalize, dim3(1), dim3(256), 0, stream, bsum, out);
}